// Net_59270548685196
// MI455X (gfx1250) — compile-verified
//
#include <hip/hip_runtime.h>
#include <math.h>

typedef __attribute__((ext_vector_type(2))) float v2f;
typedef __attribute__((ext_vector_type(8))) float v8f;

#if defined(__has_builtin)
#  if __has_builtin(__builtin_amdgcn_global_load_async_to_lds_b128)
#    define HAVE_ASYNC_LDS 1
#  endif
#endif
#ifndef HAVE_ASYNC_LDS
#  define HAVE_ASYNC_LDS 0
#endif

// Pointer types matching the builtin prototype: v4i in AS1 (global) / AS3 (LDS)
typedef int vint4 __attribute__((vector_size(16)));
typedef __attribute__((address_space(1))) vint4* gptr4;
typedef __attribute__((address_space(3))) vint4* lptr4;

// D = A(16x4 f32) * B(4x16 f32) + C(16x16 f32)  -> v_wmma_f32_16x16x4_f32
__device__ __forceinline__ v8f wmma4(v2f a, v2f b, v8f c) {
    return __builtin_amdgcn_wmma_f32_16x16x4_f32(
        /*neg_a=*/false, a, /*neg_b=*/false, b,
        /*c_mod=*/(short)0, c, /*reuse_a=*/false, /*reuse_b=*/false);
}

// Branchless order-preserving float<->u32 key (atomic scatter-max). key(finite) > 0.
__device__ __forceinline__ unsigned fkey(float f) {
    unsigned b = __float_as_uint(f);
    return b ^ (0x80000000u | (unsigned)((int)b >> 31));
}
__device__ __forceinline__ float fdec_or_zero(unsigned u) {
    // inverse: sign set -> clear sign; sign clear -> bitwise not
    unsigned dec = u ^ ((~(unsigned)((int)u >> 31)) | 0x80000000u);
    float f = __uint_as_float(dec);
    return (u != 0u) ? f : 0.0f;   // untouched slot == empty segment -> 0
}

// 16-byte global -> LDS copy; async (ASYNCcnt) when toolchain exposes the builtin.
__device__ __forceinline__ void copy16(const float* __restrict__ g, float* l) {
#if HAVE_ASYNC_LDS
    __builtin_amdgcn_global_load_async_to_lds_b128((gptr4)g, (lptr4)l, 0, 0);
#else
    *(float4*)l = *(const float4*)g;
#endif
}
__device__ __forceinline__ void stage_wait() {
#if HAVE_ASYNC_LDS
    asm volatile("s_wait_asynccnt 0" ::: "memory");
#endif
    asm volatile("s_wait_dscnt 0" ::: "memory");
}

__global__ void k_init(unsigned* __restrict__ a, unsigned* __restrict__ b, int n) {
    int i = blockIdx.x * blockDim.x + threadIdx.x;
    if (i < n) { a[i] = 0u; b[i] = 0u; }
}

// One 16-edge tile per wave; 8 waves / 256-thread block.
// message = relu(concat(feat[src](SRCD), ea) @ Wp + bp) @ Wn + bn ; scatter-max keys
// KPAD = SRCD + 4 : last float4 chunk = {edge_attr, 0, 0, 0}
template <int SRCD, int KPAD>
__global__ void k_edge(const float* __restrict__ feat, const float* __restrict__ ea,
                       const long long* __restrict__ ei,
                       const float* __restrict__ Wp, const float* __restrict__ bp,
                       const float* __restrict__ Wn, const float* __restrict__ bn,
                       unsigned* __restrict__ aggKey, int E) {
    __shared__ __align__(16) float sB[KPAD * 16];    // W_pre, zero-padded to KPAD rows
    __shared__ __align__(16) float sBn[16 * 16];     // W_net
    __shared__ float sbp[16];
    __shared__ float sbn[16];
    __shared__ __align__(16) float sA[8][16 * KPAD]; // per-wave A staging
    __shared__ __align__(16) float sM[8][16 * 16];   // per-wave relu intermediate

    const int tid = threadIdx.x;
    for (int i = tid; i < KPAD * 16; i += 256) {
        int k = i >> 4, n = i & 15;
        sB[i] = (k <= SRCD) ? Wp[k * 16 + n] : 0.0f;
    }
    for (int i = tid; i < 256; i += 256) sBn[i] = Wn[i];
    if (tid < 16) { sbp[tid] = bp[tid]; sbn[tid] = bn[tid]; }
    __syncthreads();

    const int wave = tid >> 5;
    const int lane = tid & 31;
    const int half = lane >> 4;     // 0: lanes 0-15, 1: lanes 16-31
    const int mn   = lane & 15;     // A row / B col / C col
    const int koff = half * 2;      // fragment K offset per ISA layout

    const int tile = blockIdx.x * 8 + wave;
    const int e0 = tile * 16;
    if (e0 >= E) return;

    // ---- stage A tile: lane owns row mn; halves split the float4 chunks ----
    {
        constexpr int NCH  = KPAD / 4;   // total chunks (last one = attr chunk)
        constexpr int NCH0 = NCH / 2;    // chunks owned by half 0
        int er = e0 + mn;
        if (er >= E) er = E - 1;
        const int src = (int)ei[er];
        const float* row = feat + (long long)src * SRCD;
        float* lrow = &sA[wave][mn * KPAD];
        if (half == 0) {
            #pragma unroll
            for (int c = 0; c < NCH0; ++c) copy16(row + 4 * c, lrow + 4 * c);
        } else {
            #pragma unroll
            for (int c = NCH0; c < NCH - 1; ++c) copy16(row + 4 * c, lrow + 4 * c);
            float4 t = make_float4(ea[er], 0.0f, 0.0f, 0.0f);
            *(float4*)(lrow + 4 * (NCH - 1)) = t;
        }
    }
    stage_wait();

    // ---- pre = relu(A @ Wp + bp) via WMMA f32 16x16x4 ----
    v8f acc = {0.f, 0.f, 0.f, 0.f, 0.f, 0.f, 0.f, 0.f};
    #pragma unroll
    for (int kb = 0; kb < KPAD; kb += 4) {
        v2f a, b;
        a.x = sA[wave][mn * KPAD + kb + koff];
        a.y = sA[wave][mn * KPAD + kb + koff + 1];
        b.x = sB[(kb + koff) * 16 + mn];
        b.y = sB[(kb + koff + 1) * 16 + mn];
        acc = wmma4(a, b, acc);
    }
    #pragma unroll
    for (int v = 0; v < 8; ++v) {
        int M = v + half * 8;                       // C layout: M = v (+8 upper half)
        float val = acc[v] + sbp[mn];
        sM[wave][M * 16 + mn] = fmaxf(val, 0.0f);   // relu
    }
    asm volatile("s_wait_dscnt 0" ::: "memory");

    // ---- msg = pre @ Wn + bn ----
    v8f acc2 = {0.f, 0.f, 0.f, 0.f, 0.f, 0.f, 0.f, 0.f};
    #pragma unroll
    for (int kb = 0; kb < 16; kb += 4) {
        v2f a, b;
        a.x = sM[wave][mn * 16 + kb + koff];
        a.y = sM[wave][mn * 16 + kb + koff + 1];
        b.x = sBn[(kb + koff) * 16 + mn];
        b.y = sBn[(kb + koff + 1) * 16 + mn];
        acc2 = wmma4(a, b, acc2);
    }
    // ---- scatter-max ----
    #pragma unroll
    for (int v = 0; v < 8; ++v) {
        int M = v + half * 8;
        int e = e0 + M;
        if (e < E) {
            int dst = (int)ei[E + e]; // edge_index[1]
            unsigned key = fkey(acc2[v] + sbn[mn]);
            atomicMax(&aggKey[dst * 16 + mn], key);
        }
    }
}

// h = elu(elu(decode(agg) @ Wo + bo)); one 16-node tile per wave
__global__ void k_node_mid(const unsigned* __restrict__ aggKey,
                           const float* __restrict__ Wo, const float* __restrict__ bo,
                           float* __restrict__ h, int N) {
    __shared__ __align__(16) float sB[256];
    __shared__ float sb[16];
    for (int i = threadIdx.x; i < 256; i += 256) sB[i] = Wo[i];
    if (threadIdx.x < 16) sb[threadIdx.x] = bo[threadIdx.x];
    __syncthreads();

    const int wave = threadIdx.x >> 5, lane = threadIdx.x & 31;
    const int half = lane >> 4, mn = lane & 15, koff = half * 2;
    const int tile = blockIdx.x * 8 + wave;
    const int n0 = tile * 16;
    if (n0 >= N) return;
    int nodeA = n0 + mn; if (nodeA >= N) nodeA = N - 1;

    v8f acc = {0.f, 0.f, 0.f, 0.f, 0.f, 0.f, 0.f, 0.f};
    #pragma unroll
    for (int kb = 0; kb < 16; kb += 4) {
        v2f a, b;
        uint2 u = *(const uint2*)&aggKey[nodeA * 16 + kb + koff];
        a.x = fdec_or_zero(u.x);
        a.y = fdec_or_zero(u.y);
        b.x = sB[(kb + koff) * 16 + mn];
        b.y = sB[(kb + koff + 1) * 16 + mn];
        acc = wmma4(a, b, acc);
    }
    #pragma unroll
    for (int v = 0; v < 8; ++v) {
        int node = n0 + v + half * 8;
        if (node < N) {
            float t = acc[v] + sb[mn];
            t = (t > 0.0f) ? t : expm1f(t);   // conv elu
            t = (t > 0.0f) ? t : expm1f(t);   // outer elu
            h[node * 16 + mn] = t;
        }
    }
}

// out = log_softmax(decode(agg) @ Wo + bo, axis=feature)
__global__ void k_node_out(const unsigned* __restrict__ aggKey,
                           const float* __restrict__ Wo, const float* __restrict__ bo,
                           float* __restrict__ out, int N) {
    __shared__ __align__(16) float sB[256];
    __shared__ float sb[16];
    for (int i = threadIdx.x; i < 256; i += 256) sB[i] = Wo[i];
    if (threadIdx.x < 16) sb[threadIdx.x] = bo[threadIdx.x];
    __syncthreads();

    const int wave = threadIdx.x >> 5, lane = threadIdx.x & 31;
    const int half = lane >> 4, mn = lane & 15, koff = half * 2;
    const int tile = blockIdx.x * 8 + wave;
    const int n0 = tile * 16;
    if (n0 >= N) return;
    int nodeA = n0 + mn; if (nodeA >= N) nodeA = N - 1;

    v8f acc = {0.f, 0.f, 0.f, 0.f, 0.f, 0.f, 0.f, 0.f};
    #pragma unroll
    for (int kb = 0; kb < 16; kb += 4) {
        v2f a, b;
        uint2 u = *(const uint2*)&aggKey[nodeA * 16 + kb + koff];
        a.x = fdec_or_zero(u.x);
        a.y = fdec_or_zero(u.y);
        b.x = sB[(kb + koff) * 16 + mn];
        b.y = sB[(kb + koff + 1) * 16 + mn];
        acc = wmma4(a, b, acc);
    }
    #pragma unroll
    for (int v = 0; v < 8; ++v) {
        int node = n0 + v + half * 8;
        float t = acc[v] + sb[mn];
        // feature dim lives across the 16 lanes of this half; masks < 16 stay in-half
        float mx = t;
        #pragma unroll
        for (int msk = 1; msk < 16; msk <<= 1) mx = fmaxf(mx, __shfl_xor(mx, msk, 32));
        float s = expf(t - mx);
        #pragma unroll
        for (int msk = 1; msk < 16; msk <<= 1) s += __shfl_xor(s, msk, 32);
        if (node < N) out[node * 16 + mn] = t - mx - logf(s);
    }
}

extern "C" void kernel_launch(void* const* d_in, const int* in_sizes, int n_in,
                              void* d_out, int out_size, void* d_ws, size_t ws_size,
                              hipStream_t stream) {
    const float*     x   = (const float*)d_in[0];
    const float*     ea  = (const float*)d_in[1];
    const long long* ei  = (const long long*)d_in[2];
    const float* Wp1 = (const float*)d_in[3];  const float* bp1 = (const float*)d_in[4];
    const float* Wn1 = (const float*)d_in[5];  const float* bn1 = (const float*)d_in[6];
    const float* Wo1 = (const float*)d_in[7];  const float* bo1 = (const float*)d_in[8];
    const float* Wp2 = (const float*)d_in[9];  const float* bp2 = (const float*)d_in[10];
    const float* Wn2 = (const float*)d_in[11]; const float* bn2 = (const float*)d_in[12];
    const float* Wo2 = (const float*)d_in[13]; const float* bo2 = (const float*)d_in[14];

    const int N = in_sizes[0] / 64;
    const int E = in_sizes[1];

    unsigned* agg1 = (unsigned*)d_ws;
    float*    h    = (float*)((char*)d_ws + (size_t)N * 16 * sizeof(unsigned));
    unsigned* agg2 = (unsigned*)((char*)d_ws + (size_t)N * 16 * (sizeof(unsigned) + sizeof(float)));

    const int nk = N * 16;
    k_init<<<(nk + 255) / 256, 256, 0, stream>>>(agg1, agg2, nk);

    const int etiles  = (E + 15) / 16;
    const int eblocks = (etiles + 7) / 8;
    const int ntiles  = (N + 15) / 16;
    const int nblocks = (ntiles + 7) / 8;

    k_edge<64, 68><<<eblocks, 256, 0, stream>>>(x, ea, ei, Wp1, bp1, Wn1, bn1, agg1, E);
    k_node_mid<<<nblocks, 256, 0, stream>>>(agg1, Wo1, bo1, h, N);
    k_edge<16, 20><<<eblocks, 256, 0, stream>>>(h, ea, ei, Wp2, bp2, Wn2, bn2, agg2, E);
    k_node_out<<<nblocks, 256, 0, stream>>>(agg2, Wo2, bo2, (float*)d_out, N);
}